// NGDCSDE_32933809225803
// MI455X (gfx1250) — compile-verified
//
#include <hip/hip_runtime.h>
#include <cstdint>

// ============================================================================
// NGDCSDE forward for MI455X (gfx1250, wave32, WMMA fp32 + async-LDS staging).
//
// Dead-code elimination from the reference dataflow:
//   * odeG / gruG (zv carry) never reaches an output  -> skipped entirely
//   * backward encoder GRU (h_b) unused               -> skipped
//   * post_* params unused; hv_pre_N == hv_pre (duplicated on write)
//
// All matmuls (graph hops included; [x,h] concat padded 66->80 cols so the
// DIN=2 slice rides the matrix pipe too) go through one templated WMMA GEMM:
//   * V_WMMA_F32_16X16X4_F32, full fp32 precision
//   * compile-time strides; fragments via immediate-offset ds_load_b64
//   * H tile staged with GLOBAL_LOAD_ASYNC_TO_LDS_B128 (+ s_wait_asynccnt)
//   * two accumulators to break the WMMA dependency chain
// ============================================================================

#define B_    8
#define T_    32
#define N_    512
#define S_    5
#define BN_   (B_ * N_)       // 4096
#define SBN_  (S_ * BN_)      // 20480

typedef float v2f __attribute__((ext_vector_type(2)));
typedef float v8f __attribute__((ext_vector_type(8)));

__device__ __forceinline__ v8f wmma_f32_16x16x4(v2f a, v2f b, v8f c) {
  return __builtin_amdgcn_wmma_f32_16x16x4_f32(false, a, false, b, (short)0, c,
                                               false, false);
}

__device__ __forceinline__ float sigmoid_(float x) {
  return 1.0f / (1.0f + __expf(-x));
}

// ---- compile-time layout helpers ----
constexpr int gcd_(int a, int b) { return b == 0 ? a : gcd_(b, a % b); }
// largest divisor of k that is <=128 and a multiple of 4 (512->128, 320->80)
constexpr int chunk_of(int k) {
  int c = k < 128 ? k : 128;
  while (c >= 4) { if (k % c == 0 && c % 4 == 0) return c; c -= 4; }
  return 4;
}
// padded H-tile row stride: even, gcd(stride mod 64, 64) <= 4 -> 16 rows on
// distinct bank groups for b64 fragment reads
constexpr int apad_of(int chunk) {
  int as = chunk + 4;
  while (gcd_(as % 64, 64) > 4) as += 2;
  return as;
}
// lds_w row stride: force WS % 64 == 32 so the two lane-halves read
// complementary bank sets (MBLK is a multiple of 16 -> 2*MBLK % 64 in {0,32})
constexpr int wpad_of(int mblk) {
  return 2 * mblk + (((2 * mblk) % 64) == 32 ? 0 : 32);
}

// ----------------------------------------------------------------------------
// Templated batched WMMA GEMM:  out[b] = act(H[b] @ W[b] + bias) (+ res[b])
//   H: [rows, LDH] per batch (batch stride sH); W: [KDIM,*], row stride LDW,
//   column base pre-offset by caller. MBLK columns per block, 16 rows/block,
//   wave w covers cols [16w,16w+16).
// Fragment layouts per CDNA5 ISA 7.12.2 (m/n = lane&15, half = lane>>4):
//   A 16x4 / B 4x16: VGPR r holds K = half*2 + r ; C: VGPR i holds M=i+half*8
// act: 0 none, 1 tanh, 2 sigmoid (applied before residual add)
// ----------------------------------------------------------------------------
template<int KDIM, int LDH, int LDW, int MBLK>
__global__ void gemm_t(const float* __restrict__ H, long sH,
                       const float* __restrict__ W, long sW,
                       const float* __restrict__ bias,
                       const float* __restrict__ res, long sRes, int ldres,
                       float* __restrict__ out, long sOut, int ldo, int act)
{
  constexpr int CHUNK = chunk_of(KDIM);
  constexpr int NCH   = KDIM / CHUNK;
  constexpr int AS    = apad_of(CHUNK);
  constexpr int WS    = wpad_of(MBLK);
  __shared__ float lds_a[16 * AS];
  __shared__ float lds_w[(CHUNK / 2) * WS];

  const int b    = blockIdx.y;
  const int row0 = blockIdx.x * 16;
  const float* Hb = H + (long)b * sH + (long)row0 * LDH;
  const float* Wb = W + (long)b * sW;

  const int lane = threadIdx.x & 31;
  const int wv   = threadIdx.x >> 5;
  const int n16  = lane & 15;
  const int half = lane >> 4;
  const int c0   = wv * 16;

  v8f acc0 = {0.f, 0.f, 0.f, 0.f, 0.f, 0.f, 0.f, 0.f};
  v8f acc1 = {0.f, 0.f, 0.f, 0.f, 0.f, 0.f, 0.f, 0.f};

  for (int ch = 0; ch < NCH; ++ch) {
    const int k0 = ch * CHUNK;
    // ---- H tile (16 x CHUNK): async DMA global -> LDS, b128 per lane ----
    for (int i4 = threadIdx.x; i4 < 16 * (CHUNK / 4); i4 += blockDim.x) {
      int r  = i4 / (CHUNK / 4);
      int kq = i4 - r * (CHUNK / 4);
      const float* g = Hb + (long)r * LDH + k0 + kq * 4;
      unsigned l = (unsigned)(uintptr_t)&lds_a[r * AS + kq * 4];
      asm volatile("global_load_async_to_lds_b128 %0, %1, off"
                   :: "v"(l), "v"(g) : "memory");
    }
    // ---- W tile (CHUNK x MBLK): swizzled for b64 fragment reads ----
    // element (kk,col) -> lds_w[((kk>>2)*2 + ((kk>>1)&1))*WS + col*2 + (kk&1)]
    for (int i2 = threadIdx.x; i2 < (CHUNK * MBLK) / 2; i2 += blockDim.x) {
      int kk = i2 / (MBLK / 2);
      int cp = i2 - kk * (MBLK / 2);
      float2 v = *(const float2*)(Wb + (long)(k0 + kk) * LDW + cp * 2);
      int r = (kk >> 2) * 2 + ((kk >> 1) & 1);
      int j = kk & 1;
      lds_w[r * WS + (cp * 2 + 0) * 2 + j] = v.x;
      lds_w[r * WS + (cp * 2 + 1) * 2 + j] = v.y;
    }
    if (ch + 1 < NCH) {   // speculative prefetch of the next K-chunk
      __builtin_prefetch(Wb + (long)(k0 + CHUNK + (threadIdx.x & (CHUNK - 1))) * LDW, 0, 1);
      __builtin_prefetch(Hb + (long)(threadIdx.x & 15) * LDH + k0 + CHUNK, 0, 1);
    }
    asm volatile("s_wait_asynccnt 0x0" ::: "memory");
    __syncthreads();

    const float* ap = &lds_a[n16 * AS + half * 2];
    const float* wp = &lds_w[half * WS + (c0 + n16) * 2];
#pragma unroll
    for (int k4 = 0; k4 < CHUNK / 4; k4 += 2) {   // CHUNK/4 is even everywhere
      v2f a0 = *(const v2f*)(ap + k4 * 4);
      v2f b0 = *(const v2f*)(wp + (k4 * 2) * WS);
      acc0 = wmma_f32_16x16x4(a0, b0, acc0);
      v2f a1 = *(const v2f*)(ap + (k4 + 1) * 4);
      v2f b1 = *(const v2f*)(wp + ((k4 + 1) * 2) * WS);
      acc1 = wmma_f32_16x16x4(a1, b1, acc1);
    }
    __syncthreads();
  }
  v8f acc = acc0 + acc1;

  // ---- epilogue ----
  const float bc = bias ? bias[c0 + n16] : 0.0f;
  float* ob = out + (long)b * sOut;
  const float* rb = res ? (res + (long)b * sRes) : nullptr;
#pragma unroll
  for (int i = 0; i < 8; ++i) {
    int r = row0 + i + half * 8;
    float v = acc[i] + bc;
    if (act == 1)      v = tanhf(v);
    else if (act == 2) v = sigmoid_(v);
    if (rb) v += rb[(long)r * ldres + c0 + n16];
    ob[(long)r * ldo + c0 + n16] = v;
  }
}

template<int KDIM, int LDH, int LDW, int MBLK>
static void launch_gemm(const float* H, long sH, const float* W, long sW,
                        const float* bias, const float* res, long sRes, int ldres,
                        float* out, long sOut, int ldo, int act,
                        int rowsPerBatch, int batch, hipStream_t stream)
{
  dim3 grid(rowsPerBatch / 16, batch);
  dim3 block((MBLK / 16) * 32);
  gemm_t<KDIM, LDH, LDW, MBLK><<<grid, block, 0, stream>>>(
      H, sH, W, sW, bias, res, sRes, ldres, out, sOut, ldo, act);
}

// Row-pad the GRU weights: [4*66, M] -> [4*80, M], pad rows zero.
__global__ void pad_weight(const float* __restrict__ Wsrc,
                           float* __restrict__ Wdst, int M)
{
  int idx = blockIdx.x * blockDim.x + threadIdx.x;
  if (idx >= 320 * M) return;
  int row = idx / M, c = idx - row * M;
  int hop = row / 80, rr = row - hop * 80;
  Wdst[idx] = (rr < 66) ? Wsrc[(hop * 66 + rr) * M + c] : 0.f;
}

// -------------------------- elementwise helpers -----------------------------
__global__ void bcast64(float* __restrict__ dst, const float* __restrict__ src,
                        long rows) {
  long idx = (long)blockIdx.x * blockDim.x + threadIdx.x;
  if (idx < rows * 64) dst[idx] = src[idx & 63];
}

__global__ void fill0(float* __restrict__ dst, long n) {
  long idx = (long)blockIdx.x * blockDim.x + threadIdx.x;
  if (idx < n) dst[idx] = 0.f;
}

__global__ void copy_hop0(float* __restrict__ hops, const float* __restrict__ s) {
  long idx = (long)blockIdx.x * blockDim.x + threadIdx.x;
  if (idx >= (long)BN_ * 64) return;
  long row = idx >> 6; int d = idx & 63;
  hops[row * 256 + d] = s[idx];
}

// hop0 of padded [x, h, 0] block: cols 0..1 = masked x_t, 2..65 = h, 66..79 = 0
__global__ void build_xh(float* __restrict__ hops,
                         const float* __restrict__ values,
                         const float* __restrict__ masks,
                         const float* __restrict__ h, int t)
{
  long idx = (long)blockIdx.x * blockDim.x + threadIdx.x;
  if (idx >= (long)BN_ * 80) return;
  long row = idx / 80; int dd = (int)(idx - row * 80);
  int b = (int)(row >> 9), n = (int)(row & (N_ - 1));
  float v;
  if (dd < 2) {
    long o = (((long)b * T_ + t) * N_ + n) * 2 + dd;
    v = values[o] * masks[o];
  } else if (dd < 66) {
    v = h[row * 64 + dd - 2];
  } else {
    v = 0.f;
  }
  hops[row * 320 + dd] = v;
}

// hop0 of padded [x, r*h, 0] block
__global__ void build_xrh(float* __restrict__ hops,
                          const float* __restrict__ values,
                          const float* __restrict__ masks,
                          const float* __restrict__ ru,
                          const float* __restrict__ h, int t)
{
  long idx = (long)blockIdx.x * blockDim.x + threadIdx.x;
  if (idx >= (long)BN_ * 80) return;
  long row = idx / 80; int dd = (int)(idx - row * 80);
  int b = (int)(row >> 9), n = (int)(row & (N_ - 1));
  float v;
  if (dd < 2) {
    long o = (((long)b * T_ + t) * N_ + n) * 2 + dd;
    v = values[o] * masks[o];
  } else if (dd < 66) {
    v = ru[row * 128 + dd - 2] * h[row * 64 + dd - 2];
  } else {
    v = 0.f;
  }
  hops[row * 320 + dd] = v;
}

// hv2 = u*hv1 + (1-u)*c ; hvn = obs ? hv2 : hv1 ; emit hv_pre/_N, hv_post_N
__global__ void gru_merge(const float* __restrict__ masks,
                          const float* __restrict__ hv1,
                          const float* __restrict__ ru,
                          const float* __restrict__ c,
                          float* __restrict__ hv,
                          float* __restrict__ o_pre, float* __restrict__ o_preN,
                          float* __restrict__ o_postN, int t)
{
  long idx = (long)blockIdx.x * blockDim.x + threadIdx.x;
  if (idx >= (long)BN_ * 64) return;
  long row = idx >> 6; int d = idx & 63;
  int b = (int)(row >> 9), n = (int)(row & (N_ - 1));
  long mo = (((long)b * T_ + t) * N_ + n) * 2;
  float obs = (fabsf(masks[mo]) + fabsf(masks[mo + 1]) > 1e-4f) ? 1.f : 0.f;
  float h1 = hv1[idx];
  float u  = ru[row * 128 + 64 + d];
  float h2 = u * h1 + (1.f - u) * c[idx];
  float hn = h1 * (1.f - obs) + h2 * obs;
  long oo = (((long)b * T_ + t) * N_ + n) * 64 + d;
  o_pre[oo] = h1; o_preN[oo] = h1; o_postN[oo] = hn;
  hv[idx] = hn;
}

__global__ void enc_combine(const float* __restrict__ gx,
                            const float* __restrict__ gh,
                            float* __restrict__ h, float* __restrict__ zpre_t)
{
  long idx = (long)blockIdx.x * blockDim.x + threadIdx.x;
  if (idx >= (long)BN_ * 64) return;
  long row = idx >> 6; int d = idx & 63;
  long g = row * 192 + d;
  float r  = sigmoid_(gx[g] + gh[g]);
  float z  = sigmoid_(gx[g + 64] + gh[g + 64]);
  float nn = tanhf(gx[g + 128] + r * gh[g + 128]);
  float hn = (1.f - z) * nn + z * h[idx];
  h[idx] = hn;
  zpre_t[idx] = hn;
}

__global__ void build_zin(const float* __restrict__ zpre_t,
                          const float* __restrict__ Z, float* __restrict__ zin)
{
  long idx = (long)blockIdx.x * blockDim.x + threadIdx.x;
  if (idx >= (long)SBN_ * 64) return;
  long row = idx >> 6; int d = idx & 63;
  long bn = row % BN_;
  zin[row * 128 + d]      = zpre_t[bn * 64 + d];
  zin[row * 128 + 64 + d] = Z[idx];
}

__global__ void sde_zaux(const float* __restrict__ Zp, const float* __restrict__ g1,
                         const float* __restrict__ dWt, float* __restrict__ zaux)
{
  long idx = (long)blockIdx.x * blockDim.x + threadIdx.x;
  if (idx >= (long)SBN_ * 64) return;
  zaux[idx] = Zp[idx] + g1[idx] * dWt[idx];
}

// Zn = Zp + f1 + 0.5*(g1+g2)*dW ; Zn_norm = Zn / rowmax(Zn); wave per row.
__global__ void sde_final(const float* __restrict__ Zp, const float* __restrict__ f1,
                          const float* __restrict__ g1, const float* __restrict__ g2,
                          const float* __restrict__ dWt,
                          float* __restrict__ Z, float* __restrict__ zs,
                          float* __restrict__ zsn, int t)
{
  int row  = blockIdx.x * (blockDim.x >> 5) + (threadIdx.x >> 5);
  int lane = threadIdx.x & 31;
  if (row >= SBN_) return;
  long base = (long)row * 64;
  float zn[2];
  float mx = -__builtin_inff();
#pragma unroll
  for (int j = 0; j < 2; ++j) {
    long o = base + lane + j * 32;
    float v = Zp[o] + f1[o] + 0.5f * (g1[o] + g2[o]) * dWt[o];
    zn[j] = v;
    mx = fmaxf(mx, v);
  }
  for (int off = 16; off; off >>= 1)
    mx = fmaxf(mx, __shfl_xor(mx, off, 32));
  int sb = row / N_, n = row & (N_ - 1);     // sb = s*B + b
  long zo = (((long)sb * T_ + t) * N_ + n) * 64;
#pragma unroll
  for (int j = 0; j < 2; ++j) {
    int d = lane + j * 32;
    Z[base + d]  = zn[j];
    zs[zo + d]   = zn[j];
    zsn[zo + d]  = zn[j] / mx;
  }
}

// X_pred_mean[b,tt,n,:] = hv_pre[b,t]·W[:64] + out_b + (1/S)·Σ_s Zs_N[s,b,t]·W[64:]
__global__ void out_proj(const float* __restrict__ hvpre,
                         const float* __restrict__ zsn,
                         const float* __restrict__ W,
                         const float* __restrict__ bias,
                         float* __restrict__ out)
{
  int idx = blockIdx.x * blockDim.x + threadIdx.x;
  const int TOT = B_ * (T_ - 1) * N_;
  if (idx >= TOT) return;
  int n = idx % N_;
  int rem = idx / N_;
  int tt = rem % (T_ - 1);
  int b  = rem / (T_ - 1);
  int t  = tt + 1;
  const float* hr = hvpre + (((long)b * T_ + t) * N_ + n) * 64;
  float a0 = bias[0], a1 = bias[1];
  for (int j = 0; j < 64; ++j) { float h = hr[j]; a0 += h * W[j * 2]; a1 += h * W[j * 2 + 1]; }
  float z0 = 0.f, z1 = 0.f;
  for (int s = 0; s < S_; ++s) {
    const float* zr = zsn + ((((long)s * B_ + b) * T_ + t) * N_ + n) * 64;
    for (int j = 0; j < 64; ++j) {
      float z = zr[j];
      z0 += z * W[(64 + j) * 2];
      z1 += z * W[(64 + j) * 2 + 1];
    }
  }
  out[(long)idx * 2]     = a0 + z0 * (1.f / S_);
  out[(long)idx * 2 + 1] = a1 + z1 * (1.f / S_);
}

__global__ void traj_kernel(const float* __restrict__ t_in, float* __restrict__ o) {
  int i = threadIdx.x;
  if (i < T_) o[i] = t_in[i] + 1.0f;   // DT = 1.0
}

// ============================================================================
extern "C" void kernel_launch(void* const* d_in, const int* in_sizes, int n_in,
                              void* d_out, int out_size, void* d_ws, size_t ws_size,
                              hipStream_t stream)
{
  (void)in_sizes; (void)n_in; (void)out_size; (void)ws_size;

  const float* values   = (const float*)d_in[0];
  const float* masks    = (const float*)d_in[1];
  const float* Aadj     = (const float*)d_in[2];
  const float* t_in     = (const float*)d_in[3];
  const float* dW       = (const float*)d_in[4];
  const float* gruV_Wru = (const float*)d_in[5];
  const float* gruV_bru = (const float*)d_in[6];
  const float* gruV_Wc  = (const float*)d_in[7];
  const float* gruV_bc  = (const float*)d_in[8];
  /* 9..12 gruG_* dead */
  const float* odeV_W1  = (const float*)d_in[13];
  const float* odeV_b1  = (const float*)d_in[14];
  const float* odeV_W2  = (const float*)d_in[15];
  const float* odeV_b2  = (const float*)d_in[16];
  /* 17..20 odeG_* dead */
  const float* encf_Wx  = (const float*)d_in[21];
  const float* encf_Wh  = (const float*)d_in[22];
  const float* encf_bx  = (const float*)d_in[23];
  const float* encf_bh  = (const float*)d_in[24];
  /* 25..28 enc_b_* dead */
  const float* drift_W1 = (const float*)d_in[29];
  const float* drift_b1 = (const float*)d_in[30];
  const float* drift_W2 = (const float*)d_in[31];
  const float* drift_b2 = (const float*)d_in[32];
  const float* diff_W1  = (const float*)d_in[33];
  const float* diff_b1  = (const float*)d_in[34];
  const float* diff_W2  = (const float*)d_in[35];
  const float* diff_b2  = (const float*)d_in[36];
  const float* prior_W1 = (const float*)d_in[37];
  const float* prior_b1 = (const float*)d_in[38];
  const float* prior_W2 = (const float*)d_in[39];
  const float* prior_b2 = (const float*)d_in[40];
  /* 41..44 post_* dead */
  const float* out_W    = (const float*)d_in[45];
  const float* out_b    = (const float*)d_in[46];
  const float* p_h0     = (const float*)d_in[47];
  /* 48 z0_ode dead */
  const float* p_z0sde  = (const float*)d_in[49];

  // ---- outputs (concat, return order) ----
  float* out    = (float*)d_out;
  float* o_xpm  = out;                               // [B,31,N,2]
  float* o_pre  = o_xpm  + (long)B_ * 31 * N_ * 2;   // hv_pre
  float* o_preN = o_pre  + (long)B_ * T_ * N_ * 64;  // hv_pre_N
  float* o_post = o_preN + (long)B_ * T_ * N_ * 64;  // hv_post_N
  float* o_zs   = o_post + (long)B_ * T_ * N_ * 64;  // Zs
  float* o_zsn  = o_zs   + (long)S_ * B_ * T_ * N_ * 64;
  float* o_traj = o_zsn  + (long)S_ * B_ * T_ * N_ * 64;

  // ---- workspace (~110 MB fp32) ----
  float* ws = (float*)d_ws;
  size_t off = 0;
  auto wsa = [&](size_t n) { float* p = ws + off; off += n; return p; };
  float* hv    = wsa((size_t)BN_ * 64);
  float* hv1   = wsa((size_t)BN_ * 64);
  float* h1t   = wsa((size_t)BN_ * 64);
  float* ru    = wsa((size_t)BN_ * 128);
  float* cbuf  = wsa((size_t)BN_ * 64);
  float* hopsO = wsa((size_t)BN_ * 256);   // ODE gconv concat
  float* hopsA = wsa((size_t)BN_ * 320);   // GRU gconv concat ([x,h,0] padded)
  float* hopsB = wsa((size_t)BN_ * 320);   // GRU gconv concat ([x,r*h,0])
  float* WruP  = wsa((size_t)320 * 128);   // padded gruV_Wru
  float* WcP   = wsa((size_t)320 * 64);    // padded gruV_Wc
  float* Zpre  = wsa((size_t)T_ * BN_ * 64);
  float* ench  = wsa((size_t)BN_ * 64);
  float* gx    = wsa((size_t)BN_ * 192);
  float* gh    = wsa((size_t)BN_ * 192);
  float* Zst   = wsa((size_t)SBN_ * 64);
  float* zin   = wsa((size_t)SBN_ * 128);
  float* t1    = wsa((size_t)SBN_ * 64);
  float* Zp    = wsa((size_t)SBN_ * 64);
  float* g1    = wsa((size_t)SBN_ * 64);
  float* g2    = wsa((size_t)SBN_ * 64);
  float* f1    = wsa((size_t)SBN_ * 64);
  float* zaux  = wsa((size_t)SBN_ * 64);

  const int EW  = (BN_ * 64 + 255) / 256;
  const int EWP = (BN_ * 80 + 255) / 256;
  const int EWS = (SBN_ * 64 + 255) / 256;
  const long NN = (long)N_ * N_;

  // ---- init: states + padded GRU weights ----
  bcast64<<<EW, 256, 0, stream>>>(hv, p_h0, BN_);
  bcast64<<<EWS, 256, 0, stream>>>(Zst, p_z0sde, SBN_);
  fill0<<<EW, 256, 0, stream>>>(ench, (long)BN_ * 64);
  pad_weight<<<(320 * 128 + 255) / 256, 256, 0, stream>>>(gruV_Wru, WruP, 128);
  pad_weight<<<(320 * 64 + 255) / 256, 256, 0, stream>>>(gruV_Wc, WcP, 64);

  // ==================== scan over T (odeV + gruV only) ====================
  for (int t = 0; t < T_; ++t) {
    // --- odeV layer 1: h1 = tanh(gconv(hv) @ W1 + b1)
    copy_hop0<<<EW, 256, 0, stream>>>(hopsO, hv);
    for (int k = 1; k <= 3; ++k)
      launch_gemm<512, 512, 256, 64>(Aadj, NN, hopsO + (k - 1) * 64,
                                     (long)N_ * 256, nullptr, nullptr, 0, 0,
                                     hopsO + k * 64, (long)N_ * 256, 256, 0,
                                     N_, B_, stream);
    launch_gemm<256, 256, 64, 64>(hopsO, 0, odeV_W1, 0, odeV_b1,
                                  nullptr, 0, 0, h1t, 0, 64, 1, BN_, 1, stream);
    // --- odeV layer 2: hv1 = hv + tanh(gconv(h1) @ W2 + b2)
    copy_hop0<<<EW, 256, 0, stream>>>(hopsO, h1t);
    for (int k = 1; k <= 3; ++k)
      launch_gemm<512, 512, 256, 64>(Aadj, NN, hopsO + (k - 1) * 64,
                                     (long)N_ * 256, nullptr, nullptr, 0, 0,
                                     hopsO + k * 64, (long)N_ * 256, 256, 0,
                                     N_, B_, stream);
    launch_gemm<256, 256, 64, 64>(hopsO, 0, odeV_W2, 0, odeV_b2,
                                  hv, 0, 64, hv1, 0, 64, 1, BN_, 1, stream);

    // --- gruV gates: ru = sigmoid(gconv([x,hv1]) @ Wru + bru)  (padded 80)
    build_xh<<<EWP, 256, 0, stream>>>(hopsA, values, masks, hv1, t);
    for (int k = 1; k <= 3; ++k)
      launch_gemm<512, 512, 320, 80>(Aadj, NN, hopsA + (k - 1) * 80,
                                     (long)N_ * 320, nullptr, nullptr, 0, 0,
                                     hopsA + k * 80, (long)N_ * 320, 320, 0,
                                     N_, B_, stream);
    launch_gemm<320, 320, 128, 128>(hopsA, 0, WruP, 0, gruV_bru,
                                    nullptr, 0, 0, ru, 0, 128, 2, BN_, 1, stream);
    // --- gruV candidate: c = tanh(gconv([x,r*hv1]) @ Wc + bc)
    build_xrh<<<EWP, 256, 0, stream>>>(hopsB, values, masks, ru, hv1, t);
    for (int k = 1; k <= 3; ++k)
      launch_gemm<512, 512, 320, 80>(Aadj, NN, hopsB + (k - 1) * 80,
                                     (long)N_ * 320, nullptr, nullptr, 0, 0,
                                     hopsB + k * 80, (long)N_ * 320, 320, 0,
                                     N_, B_, stream);
    launch_gemm<320, 320, 64, 64>(hopsB, 0, WcP, 0, gruV_bc,
                                  nullptr, 0, 0, cbuf, 0, 64, 1, BN_, 1, stream);
    // --- observation merge + emit hv_pre / hv_pre_N / hv_post_N
    gru_merge<<<EW, 256, 0, stream>>>(masks, hv1, ru, cbuf, hv,
                                      o_pre, o_preN, o_post, t);
  }

  // ==================== forward encoder GRU (h_b dead) ====================
  for (int t = 0; t < T_; ++t) {
    launch_gemm<64, 64, 192, 192>(o_pre + (long)t * N_ * 64, (long)T_ * N_ * 64,
                                  encf_Wx, 0, encf_bx, nullptr, 0, 0,
                                  gx, (long)N_ * 192, 192, 0, N_, B_, stream);
    launch_gemm<64, 64, 192, 192>(ench, 0, encf_Wh, 0, encf_bh,
                                  nullptr, 0, 0, gh, 0, 192, 0, BN_, 1, stream);
    enc_combine<<<EW, 256, 0, stream>>>(gx, gh, ench,
                                        Zpre + (long)t * BN_ * 64);
  }

  // ==================== SDE scan (euler_heun) ====================
  for (int t = 0; t < T_; ++t) {
    const float* dWt = dW + (long)t * SBN_ * 64;
    build_zin<<<EWS, 256, 0, stream>>>(Zpre + (long)t * BN_ * 64, Zst, zin);
    // Zp = prior MLP(zin)
    launch_gemm<128, 128, 64, 64>(zin, 0, prior_W1, 0, prior_b1,
                                  nullptr, 0, 0, t1, 0, 64, 1, SBN_, 1, stream);
    launch_gemm<64, 64, 64, 64>(t1, 0, prior_W2, 0, prior_b2,
                                nullptr, 0, 0, Zp, 0, 64, 0, SBN_, 1, stream);
    // g1 = diff MLP(Zp)
    launch_gemm<64, 64, 64, 64>(Zp, 0, diff_W1, 0, diff_b1,
                                nullptr, 0, 0, t1, 0, 64, 1, SBN_, 1, stream);
    launch_gemm<64, 64, 64, 64>(t1, 0, diff_W2, 0, diff_b2,
                                nullptr, 0, 0, g1, 0, 64, 0, SBN_, 1, stream);
    // g2 = diff MLP(Zp + g1*dW)
    sde_zaux<<<EWS, 256, 0, stream>>>(Zp, g1, dWt, zaux);
    launch_gemm<64, 64, 64, 64>(zaux, 0, diff_W1, 0, diff_b1,
                                nullptr, 0, 0, t1, 0, 64, 1, SBN_, 1, stream);
    launch_gemm<64, 64, 64, 64>(t1, 0, diff_W2, 0, diff_b2,
                                nullptr, 0, 0, g2, 0, 64, 0, SBN_, 1, stream);
    // f1 = drift MLP(Zp)
    launch_gemm<64, 64, 64, 64>(Zp, 0, drift_W1, 0, drift_b1,
                                nullptr, 0, 0, t1, 0, 64, 1, SBN_, 1, stream);
    launch_gemm<64, 64, 64, 64>(t1, 0, drift_W2, 0, drift_b2,
                                nullptr, 0, 0, f1, 0, 64, 0, SBN_, 1, stream);
    sde_final<<<SBN_ / 8, 256, 0, stream>>>(Zp, f1, g1, g2, dWt,
                                            Zst, o_zs, o_zsn, t);
  }

  // ==================== output projection + traj ====================
  out_proj<<<(B_ * (T_ - 1) * N_ + 255) / 256, 256, 0, stream>>>(
      o_pre, o_zsn, out_W, out_b, o_xpm);
  traj_kernel<<<1, 32, 0, stream>>>(t_in, o_traj);
}